// SAGE_55078660603921
// MI455X (gfx1250) — compile-verified
//
#include <hip/hip_runtime.h>

typedef __attribute__((ext_vector_type(2))) float v2f;
typedef __attribute__((ext_vector_type(8))) float v8f;

#define WMMA_F32X4(a, b, c) \
  __builtin_amdgcn_wmma_f32_16x16x4_f32(false, (a), false, (b), (short)0, (c), false, false)

// ---------------------------------------------------------------------------
// Scatter-sum: for each edge e, agg[dst[e]] += feat[src[e]], cnt[dst[e]] += 1.
// One thread handles 4 consecutive features of one edge (float4 gather,
// 4x global_atomic_add_f32). tpe_shift = log2(D/4).
// ---------------------------------------------------------------------------
__global__ void scatter_sum_kernel(const float* __restrict__ feat,
                                   const int* __restrict__ src,
                                   const int* __restrict__ dst,
                                   float* __restrict__ agg,
                                   float* __restrict__ cnt,
                                   long long total, int D, int tpe_shift) {
  long long t = (long long)blockIdx.x * blockDim.x + threadIdx.x;
  if (t >= total) return;
  const int e = (int)(t >> tpe_shift);
  const int f = (((int)t) & ((1 << tpe_shift) - 1)) << 2;
  const int s = src[e];
  const int d = dst[e];
  const float4 v = *(const float4*)(feat + (size_t)s * D + f);
  float* a = agg + (size_t)d * D + f;
  unsafeAtomicAdd(a + 0, v.x);
  unsafeAtomicAdd(a + 1, v.y);
  unsafeAtomicAdd(a + 2, v.z);
  unsafeAtomicAdd(a + 3, v.w);
  if (f == 0) unsafeAtomicAdd(cnt + d, 1.0f);
}

// ---------------------------------------------------------------------------
// Fused SAGE node update:
//   Out[i,:] = act( (Aagg[i,:]/max(cnt[i],1)) @ Wl^T + Ax[i,:] @ Wr^T + bias )
// Block = 256 threads = 8 waves. Wave w owns rows [blk.x*128 + 16w, +16) and
// columns [blk.y*64, +64) as 4 WMMA 16x16 tiles, K-loop in steps of 4 using
// V_WMMA_F32_16X16X4_F32 (full f32 precision).
//
// Fragment layouts (ISA 7.12.2):
//   A 16x4 f32 : lane L holds row (L&15); v[0..1] = K = {k0,k0+1} (L<16)
//                or {k0+2,k0+3} (L>=16)  -> float2 load at A[row][k0+2*hi]
//   B 4x16 f32 : lane L holds col (L&15); VGPR0/1 = rows {k0,k0+1}(+2 for hi)
//                -> float2 load at W[col][k0+2*hi]  (W is [N,K] row-major)
//   C/D 16x16  : v8f; VGPR v = row v (lanes 0-15) / row v+8 (lanes 16-31)
// ---------------------------------------------------------------------------
__global__ void __launch_bounds__(256)
sage_update_kernel(const float* __restrict__ Aagg,
                   const float* __restrict__ cnt,
                   const float* __restrict__ Ax,
                   const float* __restrict__ Wl,
                   const float* __restrict__ Wr,
                   const float* __restrict__ bias,
                   float* __restrict__ Out,
                   int M, int K, int N, int do_relu) {
  const int lane = threadIdx.x & 31;
  const int wave = threadIdx.x >> 5;
  const int m15  = lane & 15;
  const int hi   = lane >> 4;                 // 0: lanes 0-15, 1: lanes 16-31
  const int row_base = blockIdx.x * 128 + wave * 16;
  const int lr   = min(row_base + m15, M - 1);  // clamp: EXEC stays all-ones
  const int koff = hi * 2;
  const int ncol = blockIdx.y * 64 + m15;     // this lane's column in tile t=0

  const float inv_c = 1.0f / fmaxf(cnt[lr], 1.0f);

  const float* arow = Aagg + (size_t)lr * K + koff;
  const float* xrow = Ax   + (size_t)lr * K + koff;
  const float* wl[4];
  const float* wr[4];
#pragma unroll
  for (int t = 0; t < 4; ++t) {
    wl[t] = Wl + (size_t)(ncol + 16 * t) * K + koff;
    wr[t] = Wr + (size_t)(ncol + 16 * t) * K + koff;
  }

  v8f acc[4] = {};  // zero-init accumulators

#pragma unroll 2
  for (int k = 0; k < K; k += 4) {
    v2f aa = *(const v2f*)(arow + k);
    aa[0] *= inv_c;
    aa[1] *= inv_c;
    const v2f ax = *(const v2f*)(xrow + k);
#pragma unroll
    for (int t = 0; t < 4; ++t) {
      const v2f bl = *(const v2f*)(wl[t] + k);
      const v2f br = *(const v2f*)(wr[t] + k);
      acc[t] = WMMA_F32X4(aa, bl, acc[t]);   // mean-neighbor @ Wl^T
      acc[t] = WMMA_F32X4(ax, br, acc[t]);   // root          @ Wr^T
    }
  }

  const int out_row0 = row_base + hi * 8;
#pragma unroll
  for (int t = 0; t < 4; ++t) {
    const int col = ncol + 16 * t;
    const float bv = bias[col];
#pragma unroll
    for (int v = 0; v < 8; ++v) {
      const int r = out_row0 + v;
      if (r < M) {
        float val = acc[t][v] + bv;
        if (do_relu) val = fmaxf(val, 0.0f);
        Out[(size_t)r * N + col] = val;
      }
    }
  }
}

// ---------------------------------------------------------------------------
// Launch
// ---------------------------------------------------------------------------
extern "C" void kernel_launch(void* const* d_in, const int* in_sizes, int n_in,
                              void* d_out, int out_size, void* d_ws, size_t ws_size,
                              hipStream_t stream) {
  const float* x     = (const float*)d_in[0];
  const float* W_l0  = (const float*)d_in[1];
  const float* b_l0  = (const float*)d_in[2];
  const float* W_r0  = (const float*)d_in[3];
  const float* W_l1  = (const float*)d_in[4];
  const float* b_l1  = (const float*)d_in[5];
  const float* W_r1  = (const float*)d_in[6];
  const int* ei0_src = (const int*)d_in[7];
  const int* ei0_dst = (const int*)d_in[8];
  const int* ei1_src = (const int*)d_in[9];
  const int* ei1_dst = (const int*)d_in[10];
  float* out = (float*)d_out;
  float* ws  = (float*)d_ws;

  const long long N1 = 100000, N2 = 25000, E0 = 1600000, E1 = 400000;
  const int DIN = 128, DH = 256, DOUT = 128;

  // Workspace layout (floats): [agg0 | cnt0 | agg1 | cnt1 | h]
  float* agg0 = ws;                       // N1*DIN
  float* cnt0 = agg0 + (size_t)N1 * DIN;  // N1
  float* agg1 = cnt0 + N1;                // N2*DH
  float* cnt1 = agg1 + (size_t)N2 * DH;   // N2
  float* h    = cnt1 + N2;                // N1*DH

  // Zero the accumulation region every call (graph-capture-legal memset node).
  const size_t zero_bytes =
      ((size_t)N1 * DIN + N1 + (size_t)N2 * DH + N2) * sizeof(float);
  hipMemsetAsync(ws, 0, zero_bytes, stream);

  // Hop 0 scatter: E0 edges x 128 feats, 4 feats/thread.
  {
    const long long total = E0 * (DIN / 4);
    const int blocks = (int)((total + 255) / 256);
    scatter_sum_kernel<<<blocks, 256, 0, stream>>>(
        x, ei0_src, ei0_dst, agg0, cnt0, total, DIN, 5);
  }
  // Hop 0 update: h = relu(mean(agg0) @ Wl0^T + b0 + x @ Wr0^T)
  {
    dim3 grid((unsigned)((N1 + 127) / 128), (unsigned)(DH / 64));
    sage_update_kernel<<<grid, 256, 0, stream>>>(
        agg0, cnt0, x, W_l0, W_r0, b_l0, h, (int)N1, DIN, DH, 1);
  }
  // Hop 1 scatter: E1 edges x 256 feats.
  {
    const long long total = E1 * (DH / 4);
    const int blocks = (int)((total + 255) / 256);
    scatter_sum_kernel<<<blocks, 256, 0, stream>>>(
        h, ei1_src, ei1_dst, agg1, cnt1, total, DH, 6);
  }
  // Hop 1 update (no relu): out = mean(agg1) @ Wl1^T + b1 + h @ Wr1^T
  {
    dim3 grid((unsigned)((N2 + 127) / 128), (unsigned)(DOUT / 64));
    sage_update_kernel<<<grid, 256, 0, stream>>>(
        agg1, cnt1, h, W_l1, W_r1, b_l1, out, (int)N2, DH, DOUT, 0);
  }
}